// GATv2Layer_86208583566033
// MI455X (gfx1250) — compile-verified
//
#include <hip/hip_runtime.h>
#include <hip/hip_bf16.h>
#include <math.h>

typedef __attribute__((ext_vector_type(2))) float v2f;
typedef __attribute__((ext_vector_type(8))) float v8f;

#define IN_F   256
#define OUTC   256   // HEADS * OUT_F
#define HEADS  4
#define OUT_F  64
#define NEG_SLOPE 0.2f

__device__ __forceinline__ float leaky(float v) { return v > 0.f ? v : NEG_SLOPE * v; }

// ---------------------------------------------------------------------------
// Kernel 0: init out[i] = b[i % 256]; gmax = -inf; gsum = 0
// ---------------------------------------------------------------------------
__global__ void gat_init(float* __restrict__ out, const float* __restrict__ b,
                         float* __restrict__ gmax, float* __restrict__ gsum,
                         int outElems) {
    int i = blockIdx.x * blockDim.x + threadIdx.x;
    if (i < outElems) out[i] = b[i & (OUTC - 1)];
    if (i < HEADS) { gmax[i] = -INFINITY; gsum[i] = 0.f; }
}

// ---------------------------------------------------------------------------
// Kernel 1: h = x @ W via V_WMMA_F32_16X16X4_F32, one wave per 16x16 tile.
//   A (16x4):  lane m = lane&15, k-pair base = (lane>>4)*2  -> 2 VGPRs (v2f)
//   B (4x16):  lane n = lane&15, k-pair base = (lane>>4)*2  -> 2 VGPRs (v2f)
//   C/D 16x16: 8 VGPRs, lanes 0-15 -> M=vgpr, lanes 16-31 -> M=vgpr+8
// ---------------------------------------------------------------------------
__global__ void __launch_bounds__(256) gat_gemm_wmma(
        const float* __restrict__ x, const float* __restrict__ W,
        float* __restrict__ h, int nTilesM) {
    const int wave = blockIdx.x * (blockDim.x >> 5) + (threadIdx.x >> 5);
    const int lane = threadIdx.x & 31;
    const int mTile = wave >> 4;        // 16 n-tiles (256/16) per m-tile
    const int nTile = wave & 15;
    if (mTile >= nTilesM) return;

    const int m  = lane & 15;
    const int kb = (lane >> 4) << 1;    // 0 or 2

    const float* xrow = x + (size_t)(mTile * 16 + m) * IN_F;
    const float* wcol = W + nTile * 16 + (lane & 15);

    v8f c = {};
    #pragma unroll 4
    for (int k = 0; k < IN_F; k += 4) {
        // A fragment: two consecutive k values for this lane (8B aligned)
        v2f a = *(const v2f*)(xrow + k + kb);
        // B fragment: W is row-major (K x 256); two rows, same column
        v2f bf;
        bf.x = wcol[(size_t)(k + kb)     * OUTC];
        bf.y = wcol[(size_t)(k + kb + 1) * OUTC];
        c = __builtin_amdgcn_wmma_f32_16x16x4_f32(
                /*neg_a=*/false, a, /*neg_b=*/false, bf,
                /*c_mod=*/(short)0, c, /*reuse_a=*/false, /*reuse_b=*/false);
    }

    const int nlocal = lane & 15;
    const int rowOff = (lane >> 4) << 3;   // 0 or 8
    float* hp = h + nTile * 16 + nlocal;
    #pragma unroll
    for (int i = 0; i < 8; ++i)
        hp[(size_t)(mTile * 16 + rowOff + i) * OUTC] = c[i];
}

// ---------------------------------------------------------------------------
// Kernel 2: per-edge attention scores + global per-head max (wave per edge).
// Lane l reads elements l, l+32, ..., l+224 of the 256-wide rows (coalesced);
// head of element (l + 32*i) is i>>1 (compile-time constant per iteration).
// ---------------------------------------------------------------------------
__global__ void __launch_bounds__(256) gat_scores(
        const float* __restrict__ h, const int* __restrict__ ei,
        const float* __restrict__ att, float* __restrict__ scores,
        float* __restrict__ gmax, int E, int Nn) {
    const int wave = blockIdx.x * (blockDim.x >> 5) + (threadIdx.x >> 5);
    const int lane = threadIdx.x & 31;
    if (wave >= E + Nn) return;

    int s, d;
    if (wave < E) { s = ei[wave]; d = ei[E + wave]; }
    else          { s = d = wave - E; }

    const float* hs = h + (size_t)s * OUTC;
    const float* hd = h + (size_t)d * OUTC;

    float acc[HEADS] = {0.f, 0.f, 0.f, 0.f};
    #pragma unroll
    for (int i = 0; i < 8; ++i) {
        const int head = i >> 1;                 // (lane + 32*i) >> 6
        const int j    = lane + ((i & 1) << 5);  // (lane + 32*i) & 63
        const int idx  = lane + (i << 5);
        const float asv = att[head * 2 * OUT_F + j];
        const float adv = att[head * 2 * OUT_F + OUT_F + j];
        acc[head] += leaky(hs[idx]) * asv + leaky(hd[idx]) * adv;
    }

    #pragma unroll
    for (int hdi = 0; hdi < HEADS; ++hdi) {
        float v = acc[hdi];
        #pragma unroll
        for (int off = 16; off > 0; off >>= 1)
            v += __shfl_xor(v, off, 32);
        if (lane == 0) {
            scores[(size_t)wave * HEADS + hdi] = v;
            atomicMax(gmax + hdi, v);            // fp32 global atomic max
        }
    }
}

// ---------------------------------------------------------------------------
// Kernel 3: w = exp(score - max); per-head sum via LDS atomics -> global add.
// ---------------------------------------------------------------------------
__global__ void __launch_bounds__(256) gat_expsum(
        float* __restrict__ scores, const float* __restrict__ gmax,
        float* __restrict__ gsum, int total4) {
    __shared__ float ssum[HEADS];
    if (threadIdx.x < HEADS) ssum[threadIdx.x] = 0.f;
    __syncthreads();

    const int i = blockIdx.x * blockDim.x + threadIdx.x;
    if (i < total4) {
        const int hd = i & 3;
        const float v = __expf(scores[i] - gmax[hd]);
        scores[i] = v;
        atomicAdd(&ssum[hd], v);                 // ds_add_f32
    }
    __syncthreads();
    if (threadIdx.x < HEADS) atomicAdd(gsum + threadIdx.x, ssum[threadIdx.x]);
}

// ---------------------------------------------------------------------------
// Kernel 4: out[dst] += h[src] * (w[e,head] / sum[head])  (wave per edge)
// ---------------------------------------------------------------------------
__global__ void __launch_bounds__(256) gat_scatter(
        const float* __restrict__ h, const int* __restrict__ ei,
        const float* __restrict__ scores, const float* __restrict__ gsum,
        float* __restrict__ out, int E, int Nn) {
    const int wave = blockIdx.x * (blockDim.x >> 5) + (threadIdx.x >> 5);
    const int lane = threadIdx.x & 31;
    if (wave >= E + Nn) return;

    int s, d;
    if (wave < E) { s = ei[wave]; d = ei[E + wave]; }
    else          { s = d = wave - E; }

    float wgt[HEADS];
    #pragma unroll
    for (int hd = 0; hd < HEADS; ++hd)
        wgt[hd] = scores[(size_t)wave * HEADS + hd] / gsum[hd];

    const float* hs = h + (size_t)s * OUTC;
    float* op = out + (size_t)d * OUTC;
    #pragma unroll
    for (int i = 0; i < 8; ++i) {
        const int idx = lane + (i << 5);
        atomicAdd(op + idx, hs[idx] * wgt[i >> 1]);  // global_atomic_add_f32
    }
}

// ---------------------------------------------------------------------------
// Host-side launch
// ---------------------------------------------------------------------------
extern "C" void kernel_launch(void* const* d_in, const int* in_sizes, int n_in,
                              void* d_out, int out_size, void* d_ws, size_t ws_size,
                              hipStream_t stream) {
    const float* x   = (const float*)d_in[0];
    const float* W   = (const float*)d_in[1];
    const float* att = (const float*)d_in[2];
    const float* b   = (const float*)d_in[3];
    const int*   ei  = (const int*)d_in[4];   // edge_index (harness: int32)

    const int Nn = in_sizes[0] / IN_F;        // 20000
    const int E  = in_sizes[4] / 2;           // 320000
    const int totalEdges = E + Nn;            // edges + self loops

    float* out    = (float*)d_out;
    float* h      = (float*)d_ws;                              // Nn*256 f32
    float* scores = h + (size_t)Nn * OUTC;                     // totalEdges*4
    float* gmax   = scores + (size_t)totalEdges * HEADS;       // 4
    float* gsum   = gmax + HEADS;                              // 4

    const int outElems = Nn * OUTC;
    gat_init<<<(outElems + 255) / 256, 256, 0, stream>>>(out, b, gmax, gsum, outElems);

    const int nTilesM = Nn / 16;                      // 1250 (exact)
    const int tiles   = nTilesM * (OUTC / 16);        // 20000 tiles
    gat_gemm_wmma<<<(tiles + 7) / 8, 256, 0, stream>>>(x, W, h, nTilesM);

    gat_scores<<<(totalEdges + 7) / 8, 256, 0, stream>>>(h, ei, att, scores, gmax, E, Nn);
    gat_expsum<<<(totalEdges * HEADS + 255) / 256, 256, 0, stream>>>(
        scores, gmax, gsum, totalEdges * HEADS);
    gat_scatter<<<(totalEdges + 7) / 8, 256, 0, stream>>>(h, ei, scores, gsum, out, E, Nn);
}